// GActor_64527588655235
// MI455X (gfx1250) — compile-verified
//
#include <hip/hip_runtime.h>
#include <math.h>

typedef __attribute__((ext_vector_type(2))) float v2f;
typedef __attribute__((ext_vector_type(8))) float v8f;

// Problem constants (match reference)
static constexpr int cB  = 64;
static constexpr int cN  = 4096;
static constexpr int cE  = 32768;
static constexpr int cF  = 12;
static constexpr int cH  = 64;
static constexpr int cK1 = 3277;   // ceil(0.8*4096)
static constexpr int cK2 = 2622;   // ceil(0.8*3277)
static constexpr int TPG2 = (cK1 + 15) / 16;  // 205 row tiles per graph in conv2

// ---------------- utility fills ----------------
__global__ void fill_f32(float* p, float v, long n) {
  long t = (long)blockIdx.x * blockDim.x + threadIdx.x;
  if (t < n) p[t] = v;
}
__global__ void fill_i32(int* p, int v, long n) {
  long t = (long)blockIdx.x * blockDim.x + threadIdx.x;
  if (t < n) p[t] = v;
}

// ---------------- conv1: edge scatter (sum aggregation of raw features) ----------------
__global__ void scatter1(const float* __restrict__ x, const int* __restrict__ ei,
                         float* __restrict__ agg) {
  long t = (long)blockIdx.x * blockDim.x + threadIdx.x;
  if (t >= (long)cB * cE) return;
  int b = (int)(t >> 15);          // /E
  int e = (int)(t & (cE - 1));
  int s = ei[(long)b * 2 * cE + e];
  int d = ei[(long)b * 2 * cE + cE + e];
  const float* xs = x + ((long)b * cN + s) * cF;
  float* ag = agg + ((long)b * cN + d) * cF;
#pragma unroll
  for (int f = 0; f < cF; ++f) atomicAdd(&ag[f], xs[f]);
}

// ---------------- conv1 GEMM via f32 WMMA: h1 = relu(agg@Wr^T + b + x@Wn^T) ----------------
// One wave computes one 16x16 output tile. K=12 -> exactly 3 WMMA K-steps per matrix.
__global__ void conv1_wmma(const float* __restrict__ x, const float* __restrict__ agg,
                           const float* __restrict__ Wrel, const float* __restrict__ b1,
                           const float* __restrict__ Wroot, float* __restrict__ h1) {
  int wave = (int)(((long)blockIdx.x * blockDim.x + threadIdx.x) >> 5);
  int lane = threadIdx.x & 31;
  int ct = wave & 3;               // 4 column tiles of 16 (H=64)
  long rt = wave >> 2;             // row tile over B*N/16
  int m = lane & 15;
  int half = lane >> 4;
  long arow = rt * 16 + m;
  int n0 = ct * 16;
  const float* xr = x + arow * cF;
  const float* ar = agg + arow * cF;
  const float* wr = Wrel + (long)(n0 + m) * cF;
  const float* wn = Wroot + (long)(n0 + m) * cF;
  v8f c = {};
#pragma unroll
  for (int s = 0; s < 3; ++s) {
    int k0 = s * 4 + half * 2;     // lanes 0-15: K=0,1 ; lanes 16-31: K=2,3
    v2f a, bm;
    a.x = ar[k0];  a.y = ar[k0 + 1];
    bm.x = wr[k0]; bm.y = wr[k0 + 1];
    c = __builtin_amdgcn_wmma_f32_16x16x4_f32(false, a, false, bm, (short)0, c, false, false);
    a.x = xr[k0];  a.y = xr[k0 + 1];
    bm.x = wn[k0]; bm.y = wn[k0 + 1];
    c = __builtin_amdgcn_wmma_f32_16x16x4_f32(false, a, false, bm, (short)0, c, false, false);
  }
  float bv = b1[n0 + m];
#pragma unroll
  for (int r = 0; r < 8; ++r) {
    long row = rt * 16 + r + 8 * half;   // C/D layout: VGPR r -> M=r (+8 for lanes>=16)
    float v = c[r] + bv;
    h1[row * cH + n0 + m] = v > 0.f ? v : 0.f;
  }
}

// ---------------- scores for top-k pooling ----------------
__global__ void score1(const float* __restrict__ h1, const float* __restrict__ p,
                       float* __restrict__ sval, int* __restrict__ sidx) {
  long t = (long)blockIdx.x * blockDim.x + threadIdx.x;
  if (t >= (long)cB * cN) return;
  float pn = 0.f;
#pragma unroll
  for (int k = 0; k < cH; ++k) pn += p[k] * p[k];
  pn = sqrtf(pn);
  const float* hr = h1 + t * cH;
  float s = 0.f;
#pragma unroll
  for (int k = 0; k < cH; ++k) s += hr[k] * p[k];
  sval[t] = s / pn;
  sidx[t] = (int)(t & (cN - 1));
}

__global__ void score2(const float* __restrict__ h2, const float* __restrict__ p,
                       float* __restrict__ sval, int* __restrict__ sidx) {
  long t = (long)blockIdx.x * blockDim.x + threadIdx.x;
  if (t >= (long)cB * cN) return;
  int b = (int)(t >> 12);
  int i = (int)(t & (cN - 1));
  sidx[t] = i;
  if (i >= cK1) { sval[t] = -1e30f; return; }
  float pn = 0.f;
#pragma unroll
  for (int k = 0; k < cH; ++k) pn += p[k] * p[k];
  pn = sqrtf(pn);
  const float* hr = h2 + ((long)b * cK1 + i) * cH;
  float s = 0.f;
#pragma unroll
  for (int k = 0; k < cH; ++k) s += hr[k] * p[k];
  sval[t] = s / pn;
}

// ---------------- per-graph descending bitonic sort of (score, idx), n=4096 ----------------
__global__ __launch_bounds__(1024) void bitonic_desc(float* __restrict__ val, int* __restrict__ idx) {
  __shared__ float skey[cN];
  __shared__ int   sid[cN];
  int b = blockIdx.x;
  float* v = val + (long)b * cN;
  int*   id = idx + (long)b * cN;
  for (int i = threadIdx.x; i < cN; i += blockDim.x) { skey[i] = v[i]; sid[i] = id[i]; }
  __syncthreads();
  for (int k = 2; k <= cN; k <<= 1) {
    for (int j = k >> 1; j > 0; j >>= 1) {
      for (int t = threadIdx.x; t < cN / 2; t += blockDim.x) {
        int i = ((t & ~(j - 1)) << 1) | (t & (j - 1));
        int ixj = i | j;
        bool up = ((i & k) == 0);
        float a = skey[i], c = skey[ixj];
        if ((a < c) == up) {
          skey[i] = c; skey[ixj] = a;
          int tmp = sid[i]; sid[i] = sid[ixj]; sid[ixj] = tmp;
        }
      }
      __syncthreads();
    }
  }
  for (int i = threadIdx.x; i < cN; i += blockDim.x) { v[i] = skey[i]; id[i] = sid[i]; }
}

// ---------------- pos map + edge remap ----------------
__global__ void pos_set(const int* __restrict__ sidx, int* __restrict__ pos) {
  long t = (long)blockIdx.x * blockDim.x + threadIdx.x;
  if (t >= (long)cB * cK1) return;
  int b = (int)(t / cK1);
  int i = (int)(t % cK1);
  pos[(long)b * cN + sidx[(long)b * cN + i]] = i;
}

__global__ void remap_edges(const int* __restrict__ ei, const int* __restrict__ pos,
                            int* __restrict__ src2, int* __restrict__ dst2) {
  long t = (long)blockIdx.x * blockDim.x + threadIdx.x;
  if (t >= (long)cB * cE) return;
  int b = (int)(t >> 15);
  int e = (int)(t & (cE - 1));
  int s = ei[(long)b * 2 * cE + e];
  int d = ei[(long)b * 2 * cE + cE + e];
  int ns = pos[(long)b * cN + s];
  int nd = pos[(long)b * cN + d];
  bool ok = (ns >= 0) && (nd >= 0);
  src2[t] = ok ? ns : -1;
  dst2[t] = ok ? nd : 0;
}

// ---------------- gather + tanh gate ----------------
__global__ void gather1(const float* __restrict__ h1, const float* __restrict__ sval,
                        const int* __restrict__ sidx, float* __restrict__ hk1) {
  long t = (long)blockIdx.x * blockDim.x + threadIdx.x;
  if (t >= (long)cB * cK1 * cH) return;
  int f = (int)(t & (cH - 1));
  long ri = t >> 6;
  int b = (int)(ri / cK1);
  int r = (int)(ri % cK1);
  int perm = sidx[(long)b * cN + r];
  float g = tanhf(sval[(long)b * cN + r]);
  hk1[t] = h1[((long)b * cN + perm) * cH + f] * g;
}

__global__ void gather2(const float* __restrict__ h2, const float* __restrict__ sval,
                        const int* __restrict__ sidx, float* __restrict__ hk2) {
  long t = (long)blockIdx.x * blockDim.x + threadIdx.x;
  if (t >= (long)cB * cK2 * cH) return;
  int f = (int)(t & (cH - 1));
  long ri = t >> 6;
  int b = (int)(ri / cK2);
  int r = (int)(ri % cK2);
  int perm = sidx[(long)b * cN + r];
  float g = tanhf(sval[(long)b * cN + r]);
  hk2[t] = h2[((long)b * cK1 + perm) * cH + f] * g;
}

// ---------------- global max/mean readout (gmp || gap) ----------------
__global__ void readout(const float* __restrict__ hk, int rows, float* __restrict__ feats,
                        int accumulate) {
  int b = blockIdx.x;
  int f = threadIdx.x;  // 64 threads
  const float* base = hk + (long)b * rows * cH + f;
  float mx = -1e30f, sm = 0.f;
  for (int r = 0; r < rows; ++r) {
    float v = base[(long)r * cH];
    mx = fmaxf(mx, v);
    sm += v;
  }
  float mean = sm / (float)rows;
  if (accumulate) { feats[b * 128 + f] += mx; feats[b * 128 + 64 + f] += mean; }
  else            { feats[b * 128 + f]  = mx; feats[b * 128 + 64 + f]  = mean; }
}

// ---------------- conv2: edge scatter over pooled graph ----------------
__global__ void scatter2(const float* __restrict__ hk1, const int* __restrict__ src2,
                         const int* __restrict__ dst2, float* __restrict__ agg2) {
  long t = (long)blockIdx.x * blockDim.x + threadIdx.x;
  if (t >= (long)cB * cE * 16) return;
  int q = (int)(t & 15);
  long be = t >> 4;
  int s = src2[be];
  if (s < 0) return;
  int b = (int)(be >> 15);
  int d = dst2[be];
  const float* hs = hk1 + ((long)b * cK1 + s) * cH + q * 4;
  float* ag = agg2 + ((long)b * cK1 + d) * cH + q * 4;
#pragma unroll
  for (int f = 0; f < 4; ++f) atomicAdd(&ag[f], hs[f]);
}

// ---------------- conv2 GEMM via f32 WMMA, in-place hk1 -> h2 ----------------
// Each wave owns a full 16-row x 64-col block (4 tiles), so all reads of its rows
// complete before its writes; no cross-wave row sharing -> in-place is race-free.
__global__ void conv2_wmma(const float* __restrict__ agg2, float* __restrict__ hk1,
                           const float* __restrict__ Wrel, const float* __restrict__ b2,
                           const float* __restrict__ Wroot) {
  int wave = (int)(((long)blockIdx.x * blockDim.x + threadIdx.x) >> 5);
  int lane = threadIdx.x & 31;
  int b = wave / TPG2;
  int lt = wave % TPG2;
  int m = lane & 15;
  int half = lane >> 4;
  int rnode = lt * 16 + m;
  bool ok = rnode < cK1;
  long rsafe = ok ? rnode : 0;
  const float* aRel = agg2 + ((long)b * cK1 + rsafe) * cH;
  const float* aRoot = hk1 + ((long)b * cK1 + rsafe) * cH;
  v8f acc[4];
#pragma unroll
  for (int ct = 0; ct < 4; ++ct) {
    int n0 = ct * 16;
    const float* wr = Wrel + (long)(n0 + m) * cH;
    const float* wn = Wroot + (long)(n0 + m) * cH;
    v8f c = {};
#pragma unroll
    for (int s = 0; s < 16; ++s) {
      int k0 = s * 4 + half * 2;
      v2f a, bm;
      a.x = ok ? aRel[k0] : 0.f;
      a.y = ok ? aRel[k0 + 1] : 0.f;
      bm.x = wr[k0]; bm.y = wr[k0 + 1];
      c = __builtin_amdgcn_wmma_f32_16x16x4_f32(false, a, false, bm, (short)0, c, false, false);
      a.x = ok ? aRoot[k0] : 0.f;
      a.y = ok ? aRoot[k0 + 1] : 0.f;
      bm.x = wn[k0]; bm.y = wn[k0 + 1];
      c = __builtin_amdgcn_wmma_f32_16x16x4_f32(false, a, false, bm, (short)0, c, false, false);
    }
    acc[ct] = c;
  }
  // all reads done; now write h2 in place of hk1
#pragma unroll
  for (int ct = 0; ct < 4; ++ct) {
    int n0 = ct * 16;
    float bv = b2[n0 + m];
#pragma unroll
    for (int r = 0; r < 8; ++r) {
      int row = lt * 16 + r + 8 * half;
      if (row < cK1) {
        float v = acc[ct][r] + bv;
        hk1[((long)b * cK1 + row) * cH + n0 + m] = v > 0.f ? v : 0.f;
      }
    }
  }
}

// ---------------- heads ----------------
__global__ void head1(const float* __restrict__ feats, const float* __restrict__ W,
                      const float* __restrict__ bias, float* __restrict__ a1) {
  int t = blockIdx.x * blockDim.x + threadIdx.x;
  if (t >= cB * 64) return;
  int b = t >> 6, o = t & 63;
  float acc = bias[o];
#pragma unroll 8
  for (int k = 0; k < 128; ++k) acc += feats[b * 128 + k] * W[o * 128 + k];
  a1[t] = acc > 0.f ? acc : 0.f;
}

__global__ void head2(const float* __restrict__ a1, const float* __restrict__ W,
                      const float* __restrict__ bias, float* __restrict__ a5) {
  int t = blockIdx.x * blockDim.x + threadIdx.x;
  if (t >= cB * 10) return;
  int b = t / 10, o = t % 10;
  float acc = bias[o];
#pragma unroll 8
  for (int k = 0; k < 64; ++k) acc += a1[b * 64 + k] * W[o * 64 + k];
  a5[t] = acc;
}

// softmax across graphs (axis 0), per class; out shape [1, B, 10]
__global__ void softmax_axis0(const float* __restrict__ a5, float* __restrict__ out) {
  __shared__ float sh[cB];
  int o = blockIdx.x;      // class 0..9
  int b = threadIdx.x;     // graph 0..63
  float v = a5[b * 10 + o];
  sh[b] = v;
  __syncthreads();
  for (int s = 32; s > 0; s >>= 1) { if (b < s) sh[b] = fmaxf(sh[b], sh[b + s]); __syncthreads(); }
  float mx = sh[0];
  __syncthreads();
  float e = expf(v - mx);
  sh[b] = e;
  __syncthreads();
  for (int s = 32; s > 0; s >>= 1) { if (b < s) sh[b] += sh[b + s]; __syncthreads(); }
  out[b * 10 + o] = e / sh[0];
}

// ---------------- launch ----------------
static inline int gridFor(long n, int blk) { return (int)((n + blk - 1) / blk); }

extern "C" void kernel_launch(void* const* d_in, const int* in_sizes, int n_in,
                              void* d_out, int out_size, void* d_ws, size_t ws_size,
                              hipStream_t stream) {
  const float* x      = (const float*)d_in[0];   // [B,N,12]
  const int*   ei     = (const int*)  d_in[1];   // [B,2,E]
  const float* W1rel  = (const float*)d_in[2];   // [64,12]
  const float* b1     = (const float*)d_in[3];   // [64]
  const float* W1root = (const float*)d_in[4];   // [64,12]
  const float* p1     = (const float*)d_in[5];   // [64]
  const float* W2rel  = (const float*)d_in[6];   // [64,64]
  const float* b2     = (const float*)d_in[7];
  const float* W2root = (const float*)d_in[8];
  const float* p2     = (const float*)d_in[9];
  const float* Wa1    = (const float*)d_in[10];  // [64,128]
  const float* ba1    = (const float*)d_in[11];
  const float* Wa5    = (const float*)d_in[12];  // [10,64]
  const float* ba5    = (const float*)d_in[13];
  float* out = (float*)d_out;                    // [1,B,10] = 640 floats

  // workspace carve-up (all 4-byte elements)
  float* ws = (float*)d_ws;
  float* agg1 = ws;                                          // B*N*12
  float* h1   = agg1 + (long)cB * cN * cF;                   // B*N*64
  float* hk1  = h1   + (long)cB * cN * cH;                   // B*K1*64 (becomes h2 in-place)
  float* agg2 = hk1  + (long)cB * cK1 * cH;                  // B*K1*64 (becomes hk2)
  float* sval = agg2 + (long)cB * cK1 * cH;                  // B*N
  int*   sidx = (int*)(sval + (long)cB * cN);                // B*N
  int*   pos  = sidx + (long)cB * cN;                        // B*N
  int*   src2 = pos  + (long)cB * cN;                        // B*E
  int*   dst2 = src2 + (long)cB * cE;                        // B*E
  float* feats = (float*)(dst2 + (long)cB * cE);             // B*128
  float* a1   = feats + cB * 128;                            // B*64
  float* a5   = a1 + cB * 64;                                // B*10

  const int BLK = 256;

  // ---- GraphConv 1 ----
  fill_f32<<<gridFor((long)cB * cN * cF, BLK), BLK, 0, stream>>>(agg1, 0.f, (long)cB * cN * cF);
  scatter1<<<gridFor((long)cB * cE, BLK), BLK, 0, stream>>>(x, ei, agg1);
  // waves = (B*N/16) * 4 col tiles; 8 waves/block
  conv1_wmma<<<((long)cB * cN / 16 * 4) * 32 / BLK, BLK, 0, stream>>>(x, agg1, W1rel, b1, W1root, h1);

  // ---- TopKPooling 1 ----
  score1<<<gridFor((long)cB * cN, BLK), BLK, 0, stream>>>(h1, p1, sval, sidx);
  bitonic_desc<<<cB, 1024, 0, stream>>>(sval, sidx);
  fill_i32<<<gridFor((long)cB * cN, BLK), BLK, 0, stream>>>(pos, -1, (long)cB * cN);
  pos_set<<<gridFor((long)cB * cK1, BLK), BLK, 0, stream>>>(sidx, pos);
  remap_edges<<<gridFor((long)cB * cE, BLK), BLK, 0, stream>>>(ei, pos, src2, dst2);
  gather1<<<gridFor((long)cB * cK1 * cH, BLK), BLK, 0, stream>>>(h1, sval, sidx, hk1);
  readout<<<cB, 64, 0, stream>>>(hk1, cK1, feats, 0);

  // ---- GraphConv 2 ----
  fill_f32<<<gridFor((long)cB * cK1 * cH, BLK), BLK, 0, stream>>>(agg2, 0.f, (long)cB * cK1 * cH);
  scatter2<<<gridFor((long)cB * cE * 16, BLK), BLK, 0, stream>>>(hk1, src2, dst2, agg2);
  // waves = B * TPG2 (each wave does full 16x64 block); 4 waves/block of 128 threads
  conv2_wmma<<<(cB * TPG2) * 32 / 128, 128, 0, stream>>>(agg2, hk1, W2rel, b2, W2root);

  // ---- TopKPooling 2 (h2 lives in hk1; hk2 goes into agg2 region) ----
  score2<<<gridFor((long)cB * cN, BLK), BLK, 0, stream>>>(hk1, p2, sval, sidx);
  bitonic_desc<<<cB, 1024, 0, stream>>>(sval, sidx);
  gather2<<<gridFor((long)cB * cK2 * cH, BLK), BLK, 0, stream>>>(hk1, sval, sidx, agg2);
  readout<<<cB, 64, 0, stream>>>(agg2, cK2, feats, 1);

  // ---- heads + softmax over graphs ----
  head1<<<gridFor(cB * 64, BLK), BLK, 0, stream>>>(feats, Wa1, ba1, a1);
  head2<<<gridFor(cB * 10, BLK), BLK, 0, stream>>>(a1, Wa5, ba5, a5);
  softmax_axis0<<<10, cB, 0, stream>>>(a5, out);
}